// ExplainModule_21268678050482
// MI455X (gfx1250) — compile-verified
//
#include <hip/hip_runtime.h>
#include <hip/hip_bf16.h>

#define NN 1536
#define DD 60
#define HH 64

typedef float v2f __attribute__((ext_vector_type(2)));
typedef float v8f __attribute__((ext_vector_type(8)));

// ---------------------------------------------------------------------------
// Kernel 1: A = E @ W1[:60], B = E @ W1[60:]  via V_WMMA_F32_16X16X4_F32.
// One wave per 16x16 output tile. waves = 2 gemms * 96 mtiles * 4 ntiles = 768.
// ---------------------------------------------------------------------------
__global__ __launch_bounds__(256) void gemm_emb_w1(
    const float* __restrict__ E,    // [1536][60]
    const float* __restrict__ W1,   // [120][64]
    float* __restrict__ AB)         // [2][1536][64]
{
    const int wave = blockIdx.x * (blockDim.x >> 5) + (threadIdx.x >> 5); // 0..767
    const int lane = threadIdx.x & 31;

    const int which = wave & 1;        // 0 = row-half (W1[:60]), 1 = col-half
    const int t     = wave >> 1;       // 0..383
    const int ntile = t & 3;           // 0..3   (64 cols / 16)
    const int mtile = t >> 2;          // 0..95  (1536 rows / 16)

    const float* __restrict__ Wsub = W1 + which * (DD * HH);

    const int m    = lane & 15;        // row within tile (A) / col within tile (B)
    const int half = lane >> 4;        // ISA layout: lanes 16-31 carry K+2
    const int row  = mtile * 16 + m;
    const int n0   = ntile * 16;

    v8f acc = {};
    for (int k0 = 0; k0 < DD; k0 += 4) {
        const int ka = k0 + 2 * half;
        v2f a, b;
        // A-matrix 16x4 f32 layout (ISA 7.12.2): VGPR0 = K=ka, VGPR1 = K=ka+1
        a.x = E[row * DD + ka];
        a.y = E[row * DD + ka + 1];
        // B-matrix 4x16 f32: row striped across lanes; VGPR0 holds K=ka row
        b.x = Wsub[ka * HH + n0 + m];
        b.y = Wsub[(ka + 1) * HH + n0 + m];
        acc = __builtin_amdgcn_wmma_f32_16x16x4_f32(
            /*neg_a=*/false, a, /*neg_b=*/false, b,
            /*c_mod=*/(short)0, acc, /*reuse_a=*/false, /*reuse_b=*/false);
    }

    // D-matrix 16x16 f32: VGPR v -> row (v + 8*half), col = lane&15
    float* __restrict__ Dst = AB + which * (NN * HH);
#pragma unroll
    for (int v = 0; v < 8; ++v) {
        Dst[(mtile * 16 + v + 8 * half) * HH + n0 + m] = acc[v];
    }
}

// ---------------------------------------------------------------------------
// Kernel 2: fused gate + symmetrize + mask over 16x16 tile pairs (ti <= tj).
//   g[i,j]  = sigmoid((logit(u[i,j]) + log_alpha[i,j] + b2)/beta)
//   out[i,j]= adj[i,j] * 0.5*(g[i,j] + g[j,i])
// log_alpha[i,j] = sum_h relu(A[i,h] + B[j,h] + b1[h]) * w2[h]
// ---------------------------------------------------------------------------
__global__ __launch_bounds__(256) void gate_sym_mask(
    const float* __restrict__ Arow,   // [1536][64]
    const float* __restrict__ Bcol,   // [1536][64]
    const float* __restrict__ b1,     // [64]
    const float* __restrict__ w2,     // [64]
    const float* __restrict__ b2,     // [1]
    const int*   __restrict__ tmp,    // [1]
    const float* __restrict__ adj,    // [N][N]
    const float* __restrict__ noise,  // [N][N]
    float* __restrict__ out)          // [N][N]
{
    const int ti = blockIdx.y;
    const int tj = blockIdx.x;
    if (tj < ti) return;  // process only ti<=tj; mirrored block written here too

    // stride 68: float4-aligned (272B) and bank-conflict-free ((4c+h)%64 distinct)
    __shared__ float sAi[16][68], sBj[16][68], sAj[16][68], sBi[16][68];
    __shared__ float sB1[HH], sW2[HH];
    __shared__ float G1[16][17], G2[16][17];

    const int tid = threadIdx.x;          // 0..255
    const int r = tid >> 4;
    const int c = tid & 15;
    const int i0 = ti * 16, j0 = tj * 16;

    // Stage the four A/B row tiles: one float4 per thread per tile.
    {
        const int rr = tid >> 4;
        const int c4 = (tid & 15) * 4;
        *(float4*)&sAi[rr][c4] = *(const float4*)&Arow[(i0 + rr) * HH + c4];
        *(float4*)&sBj[rr][c4] = *(const float4*)&Bcol[(j0 + rr) * HH + c4];
        *(float4*)&sAj[rr][c4] = *(const float4*)&Arow[(j0 + rr) * HH + c4];
        *(float4*)&sBi[rr][c4] = *(const float4*)&Bcol[(i0 + rr) * HH + c4];
        if (tid < HH) { sB1[tid] = b1[tid]; sW2[tid] = w2[tid]; }
    }
    __syncthreads();

    float acc1 = 0.f, acc2 = 0.f;
#pragma unroll 8
    for (int h = 0; h < HH; ++h) {
        const float bb = sB1[h];
        const float ww = sW2[h];
        const float x1 = sAi[r][h] + sBj[c][h] + bb;  // pair (i0+r, j0+c)
        const float x2 = sAj[r][h] + sBi[c][h] + bb;  // pair (j0+r, i0+c)
        acc1 = fmaf(fmaxf(x1, 0.f), ww, acc1);
        acc2 = fmaf(fmaxf(x2, 0.f), ww, acc2);
    }

    const float beta = (float)tmp[0];
    const float bb2  = b2[0];

    const float u1 = noise[(i0 + r) * NN + (j0 + c)];
    const float z1 = (logf(u1) - log1pf(-u1) + acc1 + bb2) / beta;
    const float g1 = 1.f / (1.f + expf(-z1));

    const float u2 = noise[(j0 + r) * NN + (i0 + c)];
    const float z2 = (logf(u2) - log1pf(-u2) + acc2 + bb2) / beta;
    const float g2 = 1.f / (1.f + expf(-z2));

    G1[r][c] = g1;
    G2[r][c] = g2;
    __syncthreads();

    // out[i0+r, j0+c] = adj * 0.5*(g[i0+r,j0+c] + g[j0+c,i0+r])
    const float m1 = 0.5f * (g1 + G2[c][r]);
    out[(i0 + r) * NN + (j0 + c)] = adj[(i0 + r) * NN + (j0 + c)] * m1;

    if (ti != tj) {
        // out[j0+r, i0+c] = adj * 0.5*(g[j0+r,i0+c] + g[i0+c,j0+r])
        const float m2 = 0.5f * (g2 + G1[c][r]);
        out[(j0 + r) * NN + (i0 + c)] = adj[(j0 + r) * NN + (i0 + c)] * m2;
    }
}

// ---------------------------------------------------------------------------
// Launch
// ---------------------------------------------------------------------------
extern "C" void kernel_launch(void* const* d_in, const int* in_sizes, int n_in,
                              void* d_out, int out_size, void* d_ws, size_t ws_size,
                              hipStream_t stream) {
    const float* embed = (const float*)d_in[0];  // [1536][60]
    const float* W1    = (const float*)d_in[1];  // [120][64]
    const float* b1    = (const float*)d_in[2];  // [64]
    const float* W2    = (const float*)d_in[3];  // [64][1]
    const float* b2    = (const float*)d_in[4];  // [1]
    const float* adj   = (const float*)d_in[5];  // [N][N]
    const float* noise = (const float*)d_in[6];  // [N][N]
    const int*   tmp   = (const int*)d_in[7];    // [1]
    float* out = (float*)d_out;

    float* AB   = (float*)d_ws;        // [2][1536][64] = 786,432 bytes
    float* Arow = AB;
    float* Bcol = AB + NN * HH;

    // 768 waves total, 8 waves (256 threads) per block -> 96 blocks
    gemm_emb_w1<<<96, 256, 0, stream>>>(embed, W1, AB);

    dim3 grid(NN / 16, NN / 16);  // 96 x 96; blocks with tj<ti exit immediately
    gate_sym_mask<<<grid, 256, 0, stream>>>(Arow, Bcol, b1, W2, b2, tmp,
                                            adj, noise, out);
}